// GraphUnrollingDen_64836826301093
// MI455X (gfx1250) — compile-verified
//
#include <hip/hip_runtime.h>
#include <hip/hip_bf16.h>

// ---------------------------------------------------------------------------
// GraphUnrollingDen for MI455X (gfx1250, wave32, WMMA).
//
// conv[i,j] = [i==j]*m0 + A[i,j]*m1(i,j) + A2[i,j]*m2(i,j)
//   m_l(i,j) = b2[l] + sum_k W2[l][k] * relu(U_l[i][k] - U_l[j][k] + b1[l][k])
//   U_l      = P @ W1[l]                      (algebraic collapse of the MLP)
//   A2       = A @ A                          (fp32 WMMA 16x16x4)
//
// Pipeline: kernel1 (U), kernel2 (A2 -> d_out via WMMA), kernel3 (in-place
// elementwise conv on d_out). Workspace use: only U1/U2 (393 KB).
// ---------------------------------------------------------------------------

#define NDIM 1536
#define PDIM 32
#define FTS  32

#define BM 128
#define BN 128
#define BK 16

typedef float v2f __attribute__((ext_vector_type(2)));
typedef float v8f __attribute__((ext_vector_type(8)));

// ---------------------------------------------------------------------------
// Kernel 1: U_l = P @ W1[l]   for l = 1, 2   (bias added later, per-pair)
// ---------------------------------------------------------------------------
__global__ __launch_bounds__(256) void gud_compute_U(
    const float* __restrict__ P, const float* __restrict__ W1,
    float* __restrict__ U1, float* __restrict__ U2) {
  int idx = blockIdx.x * blockDim.x + threadIdx.x;
  if (idx >= NDIM * FTS * 2) return;
  int l   = idx / (NDIM * FTS);        // 0 -> layer 1, 1 -> layer 2
  int rem = idx - l * (NDIM * FTS);
  int i   = rem / FTS;
  int k   = rem - i * FTS;
  const float* w = W1 + (size_t)(l + 1) * PDIM * FTS;   // W1[l+1][c][k]
  const float* p = P + (size_t)i * PDIM;
  float s = 0.f;
#pragma unroll
  for (int c = 0; c < PDIM; ++c) s += p[c] * w[c * FTS + k];
  (l == 0 ? U1 : U2)[(size_t)i * FTS + k] = s;
}

// ---------------------------------------------------------------------------
// Kernel 2: C = A @ A  using V_WMMA_F32_16X16X4_F32.
// Block tile 128x128, 8 waves; each wave: 64x32 (4x2 grid of 16x16 accums).
// LDS tiles padded to kill bank conflicts on fragment reads.
// ---------------------------------------------------------------------------
__global__ __launch_bounds__(256) void gud_gemm_AA(
    const float* __restrict__ A, float* __restrict__ C) {
  __shared__ __align__(16) float As[BM][BK + 1];   // 128 x 17  (row-major)
  __shared__ __align__(16) float Bs[BK][BN + 8];   // 16  x 136 (row-major)

  const int tid  = threadIdx.x;
  const int lane = tid & 31;
  const int wave = tid >> 5;       // 0..7
  const int wm   = wave >> 2;      // 0..1 : 64-row slab
  const int wn   = wave & 3;       // 0..3 : 32-col slab

  const int tileM = blockIdx.y * BM;
  const int tileN = blockIdx.x * BN;

  v8f acc[4][2];
#pragma unroll
  for (int mi = 0; mi < 4; ++mi)
#pragma unroll
    for (int ni = 0; ni < 2; ++ni) acc[mi][ni] = {};

  const int loadR = tid >> 2;          // 0..63
  const int loadC = (tid & 3) * 4;     // 0,4,8,12
  const int bK    = tid >> 4;          // 0..15
  const int bN    = (tid & 15) * 8;    // 0..120

  for (int k0 = 0; k0 < NDIM; k0 += BK) {
    // ---- stage A tile: rows [tileM, tileM+128), cols [k0, k0+16)
#pragma unroll
    for (int p = 0; p < 2; ++p) {
      int r = loadR + p * 64;
      const float4 va = *reinterpret_cast<const float4*>(
          &A[(size_t)(tileM + r) * NDIM + k0 + loadC]);
      As[r][loadC + 0] = va.x; As[r][loadC + 1] = va.y;
      As[r][loadC + 2] = va.z; As[r][loadC + 3] = va.w;
    }
    // ---- stage B tile (B == A): rows [k0, k0+16), cols [tileN, tileN+128)
    {
      const float4 vb0 = *reinterpret_cast<const float4*>(
          &A[(size_t)(k0 + bK) * NDIM + tileN + bN]);
      const float4 vb1 = *reinterpret_cast<const float4*>(
          &A[(size_t)(k0 + bK) * NDIM + tileN + bN + 4]);
      *reinterpret_cast<float4*>(&Bs[bK][bN])     = vb0;
      *reinterpret_cast<float4*>(&Bs[bK][bN + 4]) = vb1;
    }
    __syncthreads();

    // ---- 4 WMMA k-steps of K=4 each
#pragma unroll
    for (int kk = 0; kk < BK; kk += 4) {
      const int kb   = kk + 2 * (lane >> 4);  // lanes 0-15: K=kk,kk+1; 16-31: kk+2,kk+3
      const int mrow = lane & 15;
      v2f af[4], bf[2];
#pragma unroll
      for (int mi = 0; mi < 4; ++mi) {
        int M = wm * 64 + mi * 16 + mrow;
        af[mi].x = As[M][kb];
        af[mi].y = As[M][kb + 1];
      }
#pragma unroll
      for (int ni = 0; ni < 2; ++ni) {
        int Ncol = wn * 32 + ni * 16 + mrow;
        bf[ni].x = Bs[kb][Ncol];
        bf[ni].y = Bs[kb + 1][Ncol];
      }
#pragma unroll
      for (int mi = 0; mi < 4; ++mi)
#pragma unroll
        for (int ni = 0; ni < 2; ++ni)
          acc[mi][ni] = __builtin_amdgcn_wmma_f32_16x16x4_f32(
              false, af[mi], false, bf[ni], (short)0, acc[mi][ni], false, false);
    }
    __syncthreads();
  }

  // ---- epilogue: C/D layout -> element (M = base + v + 8*(lane/16), N = lane%16)
  const int cl = lane & 15;
  const int ch = lane >> 4;
#pragma unroll
  for (int mi = 0; mi < 4; ++mi)
#pragma unroll
    for (int ni = 0; ni < 2; ++ni) {
      int Mbase = tileM + wm * 64 + mi * 16 + ch * 8;
      int Ncol  = tileN + wn * 32 + ni * 16 + cl;
#pragma unroll
      for (int v = 0; v < 8; ++v)
        C[(size_t)(Mbase + v) * NDIM + Ncol] = acc[mi][ni][v];
    }
}

// ---------------------------------------------------------------------------
// Kernel 3: in-place elementwise conv on d_out (which currently holds A2).
//   out[i,j] = A[i,j]*m1 + A2[i,j]*m2 + [i==j]*m0
// 16x16 output tile per block; U rows staged in LDS (broadcast-friendly).
// ---------------------------------------------------------------------------
__global__ __launch_bounds__(256) void gud_conv(
    const float* __restrict__ A,
    const float* __restrict__ U1, const float* __restrict__ U2,
    const float* __restrict__ b1, const float* __restrict__ W2,
    const float* __restrict__ b2, float* __restrict__ out) {
  __shared__ float sUi1[16][33], sUi2[16][33], sUj1[16][33], sUj2[16][33];
  __shared__ float sb1[2][32], sw2[2][32];

  const int tx = threadIdx.x, ty = threadIdx.y;
  const int jb = blockIdx.x * 16, ib = blockIdx.y * 16;

  sUi1[ty][tx]      = U1[(size_t)(ib + ty) * FTS + tx];
  sUi1[ty][tx + 16] = U1[(size_t)(ib + ty) * FTS + tx + 16];
  sUi2[ty][tx]      = U2[(size_t)(ib + ty) * FTS + tx];
  sUi2[ty][tx + 16] = U2[(size_t)(ib + ty) * FTS + tx + 16];
  sUj1[ty][tx]      = U1[(size_t)(jb + ty) * FTS + tx];
  sUj1[ty][tx + 16] = U1[(size_t)(jb + ty) * FTS + tx + 16];
  sUj2[ty][tx]      = U2[(size_t)(jb + ty) * FTS + tx];
  sUj2[ty][tx + 16] = U2[(size_t)(jb + ty) * FTS + tx + 16];

  const int t = ty * 16 + tx;
  if (t < 64) {
    int l = t >> 5, k = t & 31;
    sb1[l][k] = b1[(l + 1) * 32 + k];   // b1[l+1][k]
    sw2[l][k] = W2[(l + 1) * 32 + k];   // W2[l+1][k][0]
  }
  __syncthreads();

  const int i = ib + ty, j = jb + tx;
  float m1 = b2[1], m2 = b2[2];
#pragma unroll
  for (int k = 0; k < 32; ++k) {
    float t1 = sUi1[ty][k] - sUj1[tx][k] + sb1[0][k];
    m1 = __builtin_fmaf(sw2[0][k], fmaxf(t1, 0.f), m1);
    float t2 = sUi2[ty][k] - sUj2[tx][k] + sb1[1][k];
    m2 = __builtin_fmaf(sw2[1][k], fmaxf(t2, 0.f), m2);
  }

  const size_t off = (size_t)i * NDIM + j;
  float r = A[off] * m1 + out[off] * m2;   // out currently holds A2
  if (i == j) {                            // A^0 = I contribution
    float s0 = b2[0];
#pragma unroll
    for (int k = 0; k < 32; ++k) s0 = __builtin_fmaf(W2[k], fmaxf(b1[k], 0.f), s0);
    r += s0;
  }
  out[off] = r;
}

// ---------------------------------------------------------------------------
extern "C" void kernel_launch(void* const* d_in, const int* in_sizes, int n_in,
                              void* d_out, int out_size, void* d_ws, size_t ws_size,
                              hipStream_t stream) {
  (void)in_sizes; (void)n_in; (void)out_size; (void)ws_size;
  const float* A  = (const float*)d_in[0];
  const float* P  = (const float*)d_in[1];
  const float* W1 = (const float*)d_in[2];
  const float* b1 = (const float*)d_in[3];
  const float* W2 = (const float*)d_in[4];
  const float* b2 = (const float*)d_in[5];
  float* conv = (float*)d_out;

  float* U1 = (float*)d_ws;                 // 1536*32 floats
  float* U2 = U1 + (size_t)NDIM * FTS;      // 1536*32 floats  (393 KB total)

  gud_compute_U<<<(NDIM * FTS * 2 + 255) / 256, 256, 0, stream>>>(P, W1, U1, U2);

  dim3 gGemm(NDIM / BN, NDIM / BM);         // 12 x 12
  gud_gemm_AA<<<gGemm, 256, 0, stream>>>(A, conv);   // conv <- A @ A

  dim3 gConv(NDIM / 16, NDIM / 16);         // 96 x 96
  gud_conv<<<gConv, dim3(16, 16), 0, stream>>>(A, U1, U2, b1, W2, b2, conv);
}